// TrainableJointBilateralFilter_88064009437991
// MI455X (gfx1250) — compile-verified
//
#include <hip/hip_runtime.h>
#include <hip/hip_bf16.h>
#include <stdint.h>

// Joint bilateral filter, B=2, C=1, H=W=384, window 11x11 (HALF=5).
// Transcendental-bound: one v_exp_f32 per tap. Tile staged to LDS with
// CDNA5 async global->LDS copies (ASYNCcnt path).

#define HALF   5
#define WIN    11
#define TS     32                  // output tile edge
#define HT     (TS + 2 * HALF)     // 42 = tile + halo
#define LSTR   44                  // LDS row stride in float2 (padded)
#define HDIM   384
#define WDIM   384
#define LOG2E  1.44269504088896340736f

// CDNA5 async global->LDS copy: VDST = VGPR holding LDS byte address,
// VADDR = 64-bit global address. Tracked by ASYNCcnt.
__device__ __forceinline__ void async_g2l_b32(uint32_t lds_off, const float* gaddr) {
    asm volatile("global_load_async_to_lds_b32 %0, %1, off"
                 :: "v"(lds_off), "v"(gaddr)
                 : "memory");
}

__device__ __forceinline__ void wait_async_zero() {
#if __has_builtin(__builtin_amdgcn_s_wait_asynccnt)
    __builtin_amdgcn_s_wait_asynccnt(0);
#else
    asm volatile("s_wait_asynccnt 0" ::: "memory");
#endif
}

__device__ __forceinline__ float fast_exp2(float a) {
#if __has_builtin(__builtin_amdgcn_exp2f)
    return __builtin_amdgcn_exp2f(a);   // raw v_exp_f32
#else
    return exp2f(a);                    // HIP device overload (host pass sees host decl)
#endif
}

__global__ __launch_bounds__(256)
void jbf2d_kernel(const float* __restrict__ xin, const float* __restrict__ gin,
                  const float* __restrict__ psx, const float* __restrict__ psy,
                  const float* __restrict__ psc, float* __restrict__ out)
{
    __shared__ float2 tile[HT * LSTR];   // {guidance, intensity} interleaved

    const int    tid = threadIdx.x;
    const int    x0  = blockIdx.x * TS;
    const int    y0  = blockIdx.y * TS;
    const size_t img = (size_t)blockIdx.z * (HDIM * WDIM);
    const float* gimg = gin + img;
    const float* ximg = xin + img;

    // ---- Stage tile+halo into LDS. In-range cells: async DMA copies.
    // ---- Out-of-range cells: sentinel so the tap weight underflows to 0,
    // ---- reproducing the reference's border validity mask exactly.
    for (int e = tid; e < HT * HT; e += 256) {
        const int r  = e / HT;
        const int c  = e - r * HT;
        const int gy = y0 + r - HALF;
        const int gx = x0 + c - HALF;
        float2* cell = &tile[r * LSTR + c];
        if ((unsigned)gy < (unsigned)HDIM && (unsigned)gx < (unsigned)WDIM) {
            const int      off  = gy * WDIM + gx;
            const uint32_t loff = (uint32_t)(uintptr_t)cell;  // low 32b of flat LDS addr = LDS offset
            async_g2l_b32(loff,     gimg + off);  // .x = guidance
            async_g2l_b32(loff + 4, ximg + off);  // .y = intensity
        } else {
            *cell = make_float2(3.0e18f, 0.0f);   // d^2 ~ 9e36 -> exp2 -> 0, masked tap
        }
    }
    wait_async_zero();
    __syncthreads();

    // ---- Per-thread constants: exponents folded into a single exp2 per tap.
    const float sx = psx[0], sy = psy[0], sc = psc[0];
    const float kx = LOG2E / (2.0f * sx * sx);   // row-offset axis (H, sigma_x)
    const float ky = LOG2E / (2.0f * sy * sy);   // col-offset axis (W, sigma_y)
    const float kc = LOG2E / (2.0f * sc * sc);

    float ct[WIN];
#pragma unroll
    for (int i = 0; i < WIN; ++i) {
        const float d = (float)(i - HALF);
        ct[i] = -(d * d) * ky;
    }

    // ---- 4 consecutive output columns per thread: sliding 14-wide register
    // ---- window -> 154 ds_load_b64 / thread vs 484 v_exp_f32 / thread.
    const int row = tid >> 3;          // 0..31
    const int cg  = (tid & 7) * 4;     // 0,4,...,28

    float gc[4], num[4], den[4];
#pragma unroll
    for (int p = 0; p < 4; ++p) {
        gc[p]  = tile[(row + HALF) * LSTR + (cg + p + HALF)].x;
        num[p] = 0.0f;
        den[p] = 0.0f;
    }

    for (int j = 0; j < WIN; ++j) {
        const float dj    = (float)(j - HALF);
        const float rterm = -(dj * dj) * kx;
        const float2* rp  = &tile[(row + j) * LSTR + cg];

        float2 v[WIN + 3];
#pragma unroll
        for (int i = 0; i < WIN + 3; ++i) v[i] = rp[i];

#pragma unroll
        for (int p = 0; p < 4; ++p) {
#pragma unroll
            for (int i = 0; i < WIN; ++i) {
                const float d = gc[p] - v[p + i].x;
                const float w = fast_exp2(rterm + ct[i] - d * d * kc);
                num[p] = fmaf(w, v[p + i].y, num[p]);
                den[p] += w;
            }
        }
    }

    float4 res;
    res.x = num[0] / den[0];
    res.y = num[1] / den[1];
    res.z = num[2] / den[2];
    res.w = num[3] / den[3];
    float* orow = out + img + (size_t)(y0 + row) * WDIM + (x0 + cg);
    *(float4*)orow = res;   // 16B-aligned: x0+cg is a multiple of 4
}

extern "C" void kernel_launch(void* const* d_in, const int* in_sizes, int n_in,
                              void* d_out, int out_size, void* d_ws, size_t ws_size,
                              hipStream_t stream) {
    const float* xin = (const float*)d_in[0];   // input_tensor  [B,1,H,W]
    const float* gin = (const float*)d_in[1];   // guidance      [B,1,H,W]
    const float* psx = (const float*)d_in[2];   // sigma_x (1 elem)
    const float* psy = (const float*)d_in[3];   // sigma_y (1 elem)
    const float* psc = (const float*)d_in[4];   // color_sigma (1 elem)
    float* out = (float*)d_out;

    const int B = in_sizes[0] / (HDIM * WDIM);  // B*C = 2
    dim3 grid(WDIM / TS, HDIM / TS, B);         // (12, 12, 2)
    jbf2d_kernel<<<grid, dim3(256), 0, stream>>>(xin, gin, psx, psy, psc, out);
}